// LigerFusedLinearGRPOLoss_30528627540696
// MI455X (gfx1250) — compile-verified
//
#include <hip/hip_runtime.h>

typedef __attribute__((ext_vector_type(2))) float v2f;
typedef __attribute__((ext_vector_type(4))) float v4f;
typedef __attribute__((ext_vector_type(8))) float v8f;

#define B_VAL 2
#define T_VAL 1024
#define H_VAL 1024
#define V_VAL 64000
#define BETA_VAL 0.1f

#define TOK_TILE 16
#define WAVES 8
#define VT_PER_WAVE 25
#define ROWS_PER_BLOCK (WAVES * VT_PER_WAVE * 16)   // 3200 vocab rows / block
#define VCHUNKS (V_VAL / ROWS_PER_BLOCK)            // 20
#define NTOK (B_VAL * T_VAL)                        // 2048
#define TTILES (NTOK / TOK_TILE)                    // 128
#define KSTEPS (H_VAL / 4)                          // 256 WMMA K-steps
#define HP 1036   // padded LDS row stride (floats): 16B-aligned rows, bank-spread

// Online-softmax partial state for one token over a vocab subset.
struct Stat {
    float pmax;   // running max of policy logits
    int   parg;   // argmax vocab index (first occurrence on ties)
    float pse;    // sum exp(logit - pmax)
    float psum;   // plain sum of policy logits (for log_probs.mean)
    float rat;    // ref logit at current policy argmax
    float rmax;   // running max of ref logits
    float rse;    // sum exp(ref - rmax)
};

__device__ inline void stat_merge(Stat& a, const Stat& b) {
    bool bwin = (b.pmax > a.pmax) || (b.pmax == a.pmax && b.parg < a.parg);
    float nm = fmaxf(a.pmax, b.pmax);
    float se = a.pse * __expf(a.pmax - nm) + b.pse * __expf(b.pmax - nm);
    if (bwin) { a.parg = b.parg; a.rat = b.rat; }
    a.pmax = nm;
    a.pse  = se;
    a.psum += b.psum;
    float rm = fmaxf(a.rmax, b.rmax);
    a.rse  = a.rse * __expf(a.rmax - rm) + b.rse * __expf(b.rmax - rm);
    a.rmax = rm;
}

__device__ inline void stat_store(float* d, const Stat& s) {
    d[0] = s.pmax; d[1] = __int_as_float(s.parg); d[2] = s.pse; d[3] = s.psum;
    d[4] = s.rat;  d[5] = s.rmax;                 d[6] = s.rse; d[7] = 0.0f;
}

__device__ inline Stat stat_load(const float* d) {
    Stat s;
    s.pmax = d[0]; s.parg = __float_as_int(d[1]); s.pse = d[2]; s.psum = d[3];
    s.rat  = d[4]; s.rmax = d[5];                 s.rse = d[6];
    return s;
}

// ---------------------------------------------------------------------------
// Kernel 1: per-(token-tile, vocab-chunk) streaming GEMM + online softmax.
// grid = (TTILES fast, VCHUNKS slow) so concurrent blocks share a weight chunk
// in L2 (weight streamed from HBM ~once). Block = 256 threads = 8 waves.
// ---------------------------------------------------------------------------
__global__ __launch_bounds__(256)
void grpo_partial_kernel(const float* __restrict__ X, const float* __restrict__ W,
                         const float* __restrict__ XR, const float* __restrict__ WR,
                         float* __restrict__ partials)
{
    extern __shared__ float smem[];
    float* sx  = smem;                       // [16][HP] policy input tile
    float* sr  = smem + TOK_TILE * HP;       // [16][HP] ref input tile
    float* red = smem + 2 * TOK_TILE * HP;   // [8 waves][16 tok][8] merge area

    const int tid    = threadIdx.x;
    const int lane   = tid & 31;
    const int wave   = tid >> 5;
    const int l15    = lane & 15;     // token column / A row within tile
    const int half   = lane >> 4;     // K-pair selector for A/B fragments
    const int ttile  = blockIdx.x;
    const int vchunk = blockIdx.y;
    const int tok0   = ttile * TOK_TILE;

    // Cooperative load of both input tiles: iteration r loads row r fully
    // coalesced (thread tid -> float4 column tid), 16B-aligned LDS stores.
    for (int r = 0; r < TOK_TILE; ++r) {
        v4f vx = *(const v4f*)(X  + (size_t)(tok0 + r) * H_VAL + tid * 4);
        v4f vr = *(const v4f*)(XR + (size_t)(tok0 + r) * H_VAL + tid * 4);
        *(v4f*)(sx + r * HP + tid * 4) = vx;
        *(v4f*)(sr + r * HP + tid * 4) = vr;
    }
    __syncthreads();

    Stat s;
    s.pmax = -__builtin_huge_valf(); s.parg = 0x7fffffff; s.pse = 0.0f;
    s.psum = 0.0f; s.rat = 0.0f;
    s.rmax = -__builtin_huge_valf(); s.rse = 0.0f;

    // B fragment base: lane holds input[token l15][k + 2*half .. +1]
    const float* bp = sx + l15 * HP + half * 2;
    const float* br = sr + l15 * HP + half * 2;

    for (int vt = 0; vt < VT_PER_WAVE; ++vt) {
        const int vbase = vchunk * ROWS_PER_BLOCK + (vt * WAVES + wave) * 16;
        // A fragment base: lane holds weight[vbase + l15][k + 2*half .. +1]
        const float* ap = W  + (size_t)(vbase + l15) * H_VAL + half * 2;
        const float* ar = WR + (size_t)(vbase + l15) * H_VAL + half * 2;

        v8f accp = {0.f,0.f,0.f,0.f,0.f,0.f,0.f,0.f};
        v8f accr = {0.f,0.f,0.f,0.f,0.f,0.f,0.f,0.f};

        #pragma unroll 8
        for (int k4 = 0; k4 < KSTEPS; ++k4) {
            v2f a0 = *(const v2f*)(ap + 4 * k4);   // global_load_b64
            v2f a1 = *(const v2f*)(ar + 4 * k4);
            v2f b0 = *(const v2f*)(bp + 4 * k4);   // ds_load_b64
            v2f b1 = *(const v2f*)(br + 4 * k4);
            accp = __builtin_amdgcn_wmma_f32_16x16x4_f32(
                       false, a0, false, b0, (short)0, accp, false, false);
            accr = __builtin_amdgcn_wmma_f32_16x16x4_f32(
                       false, a1, false, b1, (short)0, accr, false, false);
        }

        // Online reduction: C element i = logit[vbase + i + 8*half][token l15]
        #pragma unroll
        for (int i = 0; i < 8; ++i) {
            float x = accp[i];
            float y = accr[i];
            int vrow = vbase + (half << 3) + i;
            s.psum += x;
            if (x > s.pmax) {
                s.pse = s.pse * __expf(s.pmax - x) + 1.0f;
                s.pmax = x; s.parg = vrow; s.rat = y;
            } else {
                s.pse += __expf(x - s.pmax);
            }
            if (y > s.rmax) {
                s.rse = s.rse * __expf(s.rmax - y) + 1.0f;
                s.rmax = y;
            } else {
                s.rse += __expf(y - s.rmax);
            }
        }
    }

    // Merge lane l with lane l+16 (same token, other half of vocab rows).
    Stat o;
    o.pmax = __shfl_xor(s.pmax, 16, 32);
    o.parg = __shfl_xor(s.parg, 16, 32);
    o.pse  = __shfl_xor(s.pse , 16, 32);
    o.psum = __shfl_xor(s.psum, 16, 32);
    o.rat  = __shfl_xor(s.rat , 16, 32);
    o.rmax = __shfl_xor(s.rmax, 16, 32);
    o.rse  = __shfl_xor(s.rse , 16, 32);
    stat_merge(s, o);

    if (lane < 16) stat_store(red + (wave * 16 + lane) * 8, s);
    __syncthreads();

    // One thread per token merges the 8 wave partials, writes block partial.
    if (tid < 16) {
        Stat m = stat_load(red + tid * 8);
        for (int w = 1; w < WAVES; ++w) {
            Stat b = stat_load(red + (w * 16 + tid) * 8);
            stat_merge(m, b);
        }
        stat_store(partials + ((size_t)(ttile * VCHUNKS + vchunk) * 16 + tid) * 8, m);
    }
}

// ---------------------------------------------------------------------------
// Kernel 2: merge the VCHUNKS partials per token, compute the 5 scalars.
// Single block of 256 threads; each thread owns 8 tokens.
// ---------------------------------------------------------------------------
__global__ __launch_bounds__(256)
void grpo_final_kernel(const float* __restrict__ partials,
                       const float* __restrict__ mask,
                       const float* __restrict__ rewards,
                       float* __restrict__ out)
{
    const int tid = threadIdx.x;

    float r0 = rewards[0], r1 = rewards[1];
    float rmean = 0.5f * (r0 + r1);
    // std with ddof=1 over 2 samples: sqrt(((r0-m)^2 + (r1-m)^2)/1)
    float rstd = sqrtf((r0 - rmean) * (r0 - rmean) + (r1 - rmean) * (r1 - rmean));
    float adv[2];
    adv[0] = (r0 - rmean) / (rstd + 1e-8f);
    adv[1] = (r1 - rmean) / (rstd + 1e-8f);

    float s_tok = 0.f, s_tok2 = 0.f, s_lp = 0.f;
    float s_mask0 = 0.f, s_mask1 = 0.f;
    float s_kl0 = 0.f, s_kl1 = 0.f;
    float s_pt0 = 0.f, s_pt1 = 0.f;

    for (int t = tid; t < NTOK; t += 256) {
        int tt = t >> 4, loc = t & 15;
        Stat s = stat_load(partials + ((size_t)(tt * VCHUNKS) * 16 + loc) * 8);
        for (int c = 1; c < VCHUNKS; ++c) {
            Stat b = stat_load(partials + ((size_t)(tt * VCHUNKS + c) * 16 + loc) * 8);
            stat_merge(s, b);
        }
        float lse        = s.pmax + __logf(s.pse);
        float tok_lp     = -__logf(s.pse);            // pmax - lse
        float ref_lse    = s.rmax + __logf(s.rse);
        float ref_tok_lp = s.rat - ref_lse;
        float d  = ref_tok_lp - tok_lp;
        float kl = __expf(d) - d - 1.0f;
        float m  = mask[t];
        int   b  = t / T_VAL;

        s_tok  += tok_lp;
        s_tok2 += tok_lp * tok_lp;
        s_lp   += s.psum - (float)V_VAL * lse;        // sum of log_probs over row
        float pt = (-adv[b] + BETA_VAL * kl) * m;     // ratio == 1 in forward
        if (b == 0) { s_mask0 += m; s_kl0 += kl * m; s_pt0 += pt; }
        else        { s_mask1 += m; s_kl1 += kl * m; s_pt1 += pt; }
    }

    float vals[9] = {s_tok, s_tok2, s_lp, s_mask0, s_mask1, s_kl0, s_kl1, s_pt0, s_pt1};
    #pragma unroll
    for (int off = 16; off > 0; off >>= 1)
        #pragma unroll
        for (int i = 0; i < 9; ++i) vals[i] += __shfl_xor(vals[i], off, 32);

    __shared__ float rb[8][9];
    if ((tid & 31) == 0)
        for (int i = 0; i < 9; ++i) rb[tid >> 5][i] = vals[i];
    __syncthreads();

    if (tid == 0) {
        float a[9];
        for (int i = 0; i < 9; ++i) {
            float acc = 0.f;
            for (int w = 0; w < 8; ++w) acc += rb[w][i];
            a[i] = acc;
        }
        float seql0 = fmaxf(a[3], 1.0f), seql1 = fmaxf(a[4], 1.0f);
        out[0] = 0.5f * (a[7] / seql0 + a[8] / seql1);                 // loss
        out[1] = a[0] / (float)NTOK;                                   // tok_lp.mean
        out[2] = sqrtf(fmaxf((a[1] - a[0] * a[0] / (float)NTOK) /
                             ((float)NTOK - 1.0f), 0.0f));             // std ddof=1
        out[3] = a[2] / ((float)NTOK * (float)V_VAL);                  // log_probs.mean
        out[4] = (a[5] + a[6]) / (a[3] + a[4]);                        // kl metric
    }
}

extern "C" void kernel_launch(void* const* d_in, const int* in_sizes, int n_in,
                              void* d_out, int out_size, void* d_ws, size_t ws_size,
                              hipStream_t stream) {
    const float* X  = (const float*)d_in[0];   // _input        [2,1024,1024]
    const float* W  = (const float*)d_in[1];   // weight        [64000,1024]
    const float* M  = (const float*)d_in[2];   // attention_mask[2,1024]
    const float* R  = (const float*)d_in[3];   // rewards       [2]
    const float* XR = (const float*)d_in[4];   // ref_input
    const float* WR = (const float*)d_in[5];   // ref_weight

    float* partials = (float*)d_ws;  // [TTILES][VCHUNKS][16][8] f32 ~ 1.3 MB

    size_t shmem = (size_t)(2 * TOK_TILE * HP + WAVES * 16 * 8) * sizeof(float);
    dim3 grid(TTILES, VCHUNKS);   // token tiles fast-varying -> weight L2 reuse
    grpo_partial_kernel<<<grid, 256, shmem, stream>>>(X, W, XR, WR, partials);
    grpo_final_kernel<<<1, 256, 0, stream>>>(partials, M, R, (float*)d_out);
}